// GNN_31379031065008
// MI455X (gfx1250) — compile-verified
//
#include <hip/hip_runtime.h>
#include <hip/hip_bf16.h>

typedef __attribute__((ext_vector_type(2))) float v2f;
typedef __attribute__((ext_vector_type(8))) float v8f;

#define NNODES 50000
#define NEDGES 400000
#define EMB 128
#define NLAYER 5
#define NEF 3
#define BN_EPS 1e-5f
#define MTILES 3125          // 50000 / 16

// ---------------- workspace layout (bytes, 256-aligned) ----------------
#define OFS_DEG    0                // N floats
#define OFS_AGGEA  200192           // N*3 floats
#define OFS_H      800256           // N*128 floats
#define OFS_AGG    26400256         // N*256 floats
#define OFS_Z      77600256         // N*256 floats
#define OFS_STATS  128800256        // colsum/colsq/scale/shift (4*256 floats)
#define OFS_W1T    128804352        // 256*256 floats (transposed W1, per layer)
#define OFS_W2T    129066496        // 128*256 floats (transposed W2, per layer)
// total ~129.2 MB

// ---------------------------------------------------------------------
// Precompute: deg[n] = 1 + #incoming edges; agg_ea[n] = onehot(1) + sum ea
// ---------------------------------------------------------------------
__global__ void k_init_node(float* __restrict__ deg, float* __restrict__ aggea) {
    int n = blockIdx.x * 256 + threadIdx.x;
    if (n < NNODES) {
        deg[n] = 1.0f;                       // self loop
        aggea[n * 3 + 0] = 0.0f;
        aggea[n * 3 + 1] = 1.0f;             // self_loop_attr col -2 == col 1
        aggea[n * 3 + 2] = 0.0f;
    }
}

__global__ void k_scatter_ea(const int* __restrict__ ei, const float* __restrict__ ea,
                             float* __restrict__ deg, float* __restrict__ aggea) {
    int e = blockIdx.x * 256 + threadIdx.x;
    if (e < NEDGES) {
        int d = ei[NEDGES + e];              // edge_index[1][e]
        atomicAdd(&aggea[d * 3 + 0], ea[e * 3 + 0]);
        atomicAdd(&aggea[d * 3 + 1], ea[e * 3 + 1]);
        atomicAdd(&aggea[d * 3 + 2], ea[e * 3 + 2]);
        atomicAdd(&deg[d], 1.0f);
    }
}

// h0 = x @ in_w + in_b   ([N,3] @ [3,128])
__global__ void k_input(const float* __restrict__ x, const float* __restrict__ inw,
                        const float* __restrict__ inb, float* __restrict__ h) {
    int t = blockIdx.x * 256 + threadIdx.x;
    if (t < NNODES * EMB) {
        int n = t >> 7, c = t & (EMB - 1);
        h[t] = fmaf(x[n * 3 + 0], inw[c],
               fmaf(x[n * 3 + 1], inw[EMB + c],
               fmaf(x[n * 3 + 2], inw[2 * EMB + c], inb[c])));
    }
}

// wt[n*K + k] = w[k*N + n]
__global__ void k_transpose(const float* __restrict__ w, float* __restrict__ wt,
                            int K, int N) {
    int t = blockIdx.x * 256 + threadIdx.x;
    if (t < K * N) {
        int k = t / N, n = t - k * N;
        wt[n * K + k] = w[t];
    }
}

// agg[:,0:128] = h (self-loop part of agg_x); agg[:,128:256] = agg_ea@edge_w + deg*edge_b
__global__ void k_prep_agg(const float* __restrict__ h, const float* __restrict__ aggea,
                           const float* __restrict__ deg, const float* __restrict__ ew,
                           const float* __restrict__ eb, float* __restrict__ agg) {
    int t = blockIdx.x * 256 + threadIdx.x;
    if (t < NNODES * 2 * EMB) {
        int n = t >> 8, c = t & 255;
        float v;
        if (c < EMB) {
            v = h[n * EMB + c];
        } else {
            int cc = c - EMB;
            v = fmaf(aggea[n * 3 + 0], ew[cc],
                fmaf(aggea[n * 3 + 1], ew[EMB + cc],
                fmaf(aggea[n * 3 + 2], ew[2 * EMB + cc], deg[n] * eb[cc])));
        }
        agg[t] = v;
    }
}

// One wave per edge: lane handles 4 channels (float4 gather, 4 f32 atomics)
__global__ void k_scatter_h(const int* __restrict__ ei, const float* __restrict__ h,
                            float* __restrict__ agg) {
    int gid = blockIdx.x * 256 + threadIdx.x;
    int e = gid >> 5;
    int lane = gid & 31;
    if (e < NEDGES) {
        int s = ei[e];
        int d = ei[NEDGES + e];
        float4 hv = reinterpret_cast<const float4*>(h + (size_t)s * EMB)[lane];
        float* o = agg + (size_t)d * 2 * EMB + lane * 4;
        atomicAdd(o + 0, hv.x);
        atomicAdd(o + 1, hv.y);
        atomicAdd(o + 2, hv.z);
        atomicAdd(o + 3, hv.w);
    }
}

__global__ void k_zero(float* __restrict__ p) {
    p[blockIdx.x * 256 + threadIdx.x] = 0.0f;
}

// ---------------------------------------------------------------------
// z = agg @ W1 + b1  (M=50000, K=256, N=256) via V_WMMA_F32_16X16X4_F32
// Wave tile: 16(M) x 64(N) -> 4 accumulators, A float2 reused across 4 WMMAs.
// Block: 512 thr = 16 waves = 4 M-tiles x 4 N-waves. B taken from W1t[n][k]
// so each lane's (k,k+1) pair is one b64 load. Also accumulates column
// sum/sumsq for batch norm.
// ---------------------------------------------------------------------
__global__ void __launch_bounds__(512) k_gemm1(const float* __restrict__ agg,
        const float* __restrict__ w1t, const float* __restrict__ b1,
        float* __restrict__ z, float* __restrict__ colsum, float* __restrict__ colsq) {
    int wave = threadIdx.x >> 5;                 // 0..15
    int lane = threadIdx.x & 31;
    int mt = blockIdx.x * 4 + (wave >> 2);
    if (mt >= MTILES) return;                    // wave-uniform: EXEC stays all-1
    int nt = wave & 3;
    int m0 = mt << 4, n0 = nt << 6;              // 64 columns per wave
    int half = lane >> 4, l16 = lane & 15;

    // A: lane l16 = row m0+l16; vgpr pair = K {half*2, half*2+1}
    const float* abase = agg + (size_t)(m0 + l16) * 256 + half * 2;
    // B: lane l16 = col n0+16j+l16; W1t row-contiguous in k
    const float* bbase = w1t + (size_t)(n0 + l16) * 256 + half * 2;

    v8f acc0 = {}, acc1 = {}, acc2 = {}, acc3 = {};
    #pragma unroll 4
    for (int k0 = 0; k0 < 256; k0 += 4) {
        v2f a  = *reinterpret_cast<const v2f*>(abase + k0);
        v2f b0 = *reinterpret_cast<const v2f*>(bbase + k0);
        v2f b1v = *reinterpret_cast<const v2f*>(bbase + 16 * 256 + k0);
        v2f b2v = *reinterpret_cast<const v2f*>(bbase + 32 * 256 + k0);
        v2f b3v = *reinterpret_cast<const v2f*>(bbase + 48 * 256 + k0);
        acc0 = __builtin_amdgcn_wmma_f32_16x16x4_f32(false, a, false, b0,  (short)0, acc0, false, false);
        acc1 = __builtin_amdgcn_wmma_f32_16x16x4_f32(false, a, false, b1v, (short)0, acc1, false, false);
        acc2 = __builtin_amdgcn_wmma_f32_16x16x4_f32(false, a, false, b2v, (short)0, acc2, false, false);
        acc3 = __builtin_amdgcn_wmma_f32_16x16x4_f32(false, a, false, b3v, (short)0, acc3, false, false);
    }

    v8f accs[4] = {acc0, acc1, acc2, acc3};
    #pragma unroll
    for (int j = 0; j < 4; ++j) {
        int col = n0 + j * 16 + l16;
        float bias = b1[col];
        float s = 0.0f, sq = 0.0f;
        #pragma unroll
        for (int r = 0; r < 8; ++r) {
            float v = accs[j][r] + bias;         // D: vgpr r, half -> M = m0 + half*8 + r
            z[(size_t)(m0 + half * 8 + r) * 256 + col] = v;
            s += v;
            sq += v * v;
        }
        atomicAdd(&colsum[col], s);
        atomicAdd(&colsq[col], sq);
    }
}

// fold BN into scale/shift: y = scale*z + shift, then ReLU in gemm2
__global__ void k_bn_final(const float* __restrict__ colsum, const float* __restrict__ colsq,
                           const float* __restrict__ g, const float* __restrict__ bb,
                           float* __restrict__ scale, float* __restrict__ shift) {
    int c = threadIdx.x;                 // 256 threads
    float mean = colsum[c] * (1.0f / NNODES);
    float var  = colsq[c] * (1.0f / NNODES) - mean * mean;   // biased var (ddof=0)
    float sc   = g[c] * rsqrtf(var + BN_EPS);
    scale[c] = sc;
    shift[c] = bb[c] - mean * sc;
}

// ---------------------------------------------------------------------
// h_next = relu?( relu(scale*z+shift) @ W2 + b2 )  (M=50000, K=256, N=128)
// Wave tile 16x64 -> 2 N-waves per M-tile; block = 16 waves = 8 M-tiles.
// BN + ReLU fused into the A-operand load (computed once per wave, reused
// across 4 WMMAs).
// ---------------------------------------------------------------------
__global__ void __launch_bounds__(512) k_gemm2(const float* __restrict__ z,
        const float* __restrict__ scale, const float* __restrict__ shift,
        const float* __restrict__ w2t, const float* __restrict__ b2,
        float* __restrict__ hout, int do_relu) {
    int wave = threadIdx.x >> 5;                 // 0..15
    int lane = threadIdx.x & 31;
    int mt = blockIdx.x * 8 + (wave >> 1);
    if (mt >= MTILES) return;                    // wave-uniform guard
    int nt = wave & 1;
    int m0 = mt << 4, n0 = nt << 6;
    int half = lane >> 4, l16 = lane & 15;

    const float* abase = z + (size_t)(m0 + l16) * 256 + half * 2;
    const float* bbase = w2t + (size_t)(n0 + l16) * 256 + half * 2;
    const float* scb = scale + half * 2;
    const float* shb = shift + half * 2;

    v8f acc0 = {}, acc1 = {}, acc2 = {}, acc3 = {};
    #pragma unroll 4
    for (int k0 = 0; k0 < 256; k0 += 4) {
        v2f zr = *reinterpret_cast<const v2f*>(abase + k0);
        v2f sv = *reinterpret_cast<const v2f*>(scb + k0);
        v2f hv = *reinterpret_cast<const v2f*>(shb + k0);
        v2f a;
        a.x = fmaxf(fmaf(zr.x, sv.x, hv.x), 0.0f);
        a.y = fmaxf(fmaf(zr.y, sv.y, hv.y), 0.0f);
        v2f b0 = *reinterpret_cast<const v2f*>(bbase + k0);
        v2f b1v = *reinterpret_cast<const v2f*>(bbase + 16 * 256 + k0);
        v2f b2v = *reinterpret_cast<const v2f*>(bbase + 32 * 256 + k0);
        v2f b3v = *reinterpret_cast<const v2f*>(bbase + 48 * 256 + k0);
        acc0 = __builtin_amdgcn_wmma_f32_16x16x4_f32(false, a, false, b0,  (short)0, acc0, false, false);
        acc1 = __builtin_amdgcn_wmma_f32_16x16x4_f32(false, a, false, b1v, (short)0, acc1, false, false);
        acc2 = __builtin_amdgcn_wmma_f32_16x16x4_f32(false, a, false, b2v, (short)0, acc2, false, false);
        acc3 = __builtin_amdgcn_wmma_f32_16x16x4_f32(false, a, false, b3v, (short)0, acc3, false, false);
    }

    v8f accs[4] = {acc0, acc1, acc2, acc3};
    #pragma unroll
    for (int j = 0; j < 4; ++j) {
        int col = n0 + j * 16 + l16;
        float bias = b2[col];
        #pragma unroll
        for (int r = 0; r < 8; ++r) {
            float v = accs[j][r] + bias;
            if (do_relu) v = fmaxf(v, 0.0f);
            hout[(size_t)(m0 + half * 8 + r) * EMB + col] = v;
        }
    }
}

// ---------------------------------------------------------------------
extern "C" void kernel_launch(void* const* d_in, const int* in_sizes, int n_in,
                              void* d_out, int out_size, void* d_ws, size_t ws_size,
                              hipStream_t stream) {
    const float* x      = (const float*)d_in[0];
    const int*   ei     = (const int*)  d_in[1];
    const float* ea     = (const float*)d_in[2];
    const float* in_w   = (const float*)d_in[3];
    const float* in_b   = (const float*)d_in[4];
    const float* edge_w = (const float*)d_in[5];   // [L,3,128]
    const float* edge_b = (const float*)d_in[6];   // [L,128]
    const float* mlp_w1 = (const float*)d_in[7];   // [L,256,256]
    const float* mlp_b1 = (const float*)d_in[8];   // [L,256]
    const float* bn_g   = (const float*)d_in[9];   // [L,256]
    const float* bn_b   = (const float*)d_in[10];  // [L,256]
    const float* mlp_w2 = (const float*)d_in[11];  // [L,256,128]
    const float* mlp_b2 = (const float*)d_in[12];  // [L,128]
    float* out = (float*)d_out;

    char* ws = (char*)d_ws;
    float* deg    = (float*)(ws + OFS_DEG);
    float* aggea  = (float*)(ws + OFS_AGGEA);
    float* h      = (float*)(ws + OFS_H);
    float* agg    = (float*)(ws + OFS_AGG);
    float* z      = (float*)(ws + OFS_Z);
    float* colsum = (float*)(ws + OFS_STATS);          // 256
    float* colsq  = colsum + 256;                      // 256
    float* scale  = colsum + 512;                      // 256
    float* shift  = colsum + 768;                      // 256
    float* w1t    = (float*)(ws + OFS_W1T);            // [256,256] (n-major)
    float* w2t    = (float*)(ws + OFS_W2T);            // [128,256] (n-major)

    // one-time edge-attr aggregation (commutes with the per-layer linear map)
    k_init_node <<<(NNODES + 255) / 256, 256, 0, stream>>>(deg, aggea);
    k_scatter_ea<<<(NEDGES + 255) / 256, 256, 0, stream>>>(ei, ea, deg, aggea);
    // input embedding
    k_input<<<NNODES * EMB / 256, 256, 0, stream>>>(x, in_w, in_b, h);

    for (int l = 0; l < NLAYER; ++l) {
        k_prep_agg<<<NNODES * 2 * EMB / 256, 256, 0, stream>>>(
            h, aggea, deg, edge_w + (size_t)l * NEF * EMB, edge_b + (size_t)l * EMB, agg);
        k_scatter_h<<<NEDGES * 32 / 256, 256, 0, stream>>>(ei, h, agg);
        k_zero<<<2, 256, 0, stream>>>(colsum);          // zero colsum + colsq
        k_transpose<<<256 * 256 / 256, 256, 0, stream>>>(
            mlp_w1 + (size_t)l * 256 * 256, w1t, 256, 256);
        k_transpose<<<256 * 128 / 256, 256, 0, stream>>>(
            mlp_w2 + (size_t)l * 256 * EMB, w2t, 256, 128);
        k_gemm1<<<(MTILES + 3) / 4, 512, 0, stream>>>(
            agg, w1t, mlp_b1 + (size_t)l * 256, z, colsum, colsq);
        k_bn_final<<<1, 256, 0, stream>>>(colsum, colsq,
            bn_g + (size_t)l * 256, bn_b + (size_t)l * 256, scale, shift);
        float* hout = (l == NLAYER - 1) ? out : h;
        k_gemm2<<<(MTILES + 7) / 8, 512, 0, stream>>>(
            z, scale, shift, w2t, mlp_b2 + (size_t)l * EMB,
            hout, (l < NLAYER - 1) ? 1 : 0);
    }
}